// CSRConverterOriginal_61933428414708
// MI455X (gfx1250) — compile-verified
//
#include <hip/hip_runtime.h>
#include <hip/hip_bf16.h>

// Dense -> CSR compaction for a 4096x8192 f32 matrix (~30% nnz).
// Memory-bound: ~385-512 MiB of HBM traffic => ~16-22 us at 23.3 TB/s.
// MI455X-specific choices:
//  * 192 MB L2 holds the whole 128 MiB input; pass order
//    count -> scan -> compact -> fill plus non-temporal output stores keeps
//    x resident in L2 for the second read.
//  * wave32 shuffle scans (gfx1250 is wave32-only), 8 waves per 256-thd block.
//  * compact stages each row's CSR segment in LDS (48 KB of the 320 KB WGP
//    LDS) and flushes it with dense full-wave stores; the value flush uses
//    GLOBAL_STORE_ASYNC_FROM_LDS_B32 (ASYNCcnt) — builtin confirmed present.
//  * global_prefetch_b8 for the next row tile in the compaction loop.

#define ROWS   4096
#define COLS   8192
#define NUMEL  ((size_t)ROWS * (size_t)COLS)

#define HAVE_ASYNC_LDS_STORE 0
#if defined(__has_builtin)
#if __has_builtin(__builtin_amdgcn_global_store_async_from_lds_b32) && \
    __has_builtin(__builtin_amdgcn_s_wait_asynccnt)
#undef HAVE_ASYNC_LDS_STORE
#define HAVE_ASYNC_LDS_STORE 1
#endif
#endif

typedef __attribute__((address_space(1))) int gas_int;  // global (AS1) int
typedef __attribute__((address_space(3))) int las_int;  // LDS    (AS3) int

// ---------------------------------------------------------------------------
// Kernel 1: per-row nonzero counts. One block per row, 256 threads,
// float4 (global_load_b128) loads, wave32 xor-reduce + LDS reduce.
// ---------------------------------------------------------------------------
__global__ void __launch_bounds__(256)
count_rows(const float* __restrict__ x, int* __restrict__ counts) {
    const int row  = blockIdx.x;
    const int tid  = threadIdx.x;
    const int lane = tid & 31;
    const int wv   = tid >> 5;

    const float4* __restrict__ rowp =
        (const float4*)(x + (size_t)row * COLS);   // 2048 float4 per row

    int c = 0;
#pragma unroll
    for (int it = 0; it < 8; ++it) {
        float4 v = rowp[it * 256 + tid];
        c += (v.x != 0.0f) + (v.y != 0.0f) + (v.z != 0.0f) + (v.w != 0.0f);
    }
    // wave32 reduction
#pragma unroll
    for (int off = 16; off > 0; off >>= 1) c += __shfl_xor(c, off, 32);

    __shared__ int wsum[8];
    if (lane == 0) wsum[wv] = c;
    __syncthreads();
    if (tid == 0) {
        int s = 0;
#pragma unroll
        for (int i = 0; i < 8; ++i) s += wsum[i];
        counts[row] = s;
    }
}

// ---------------------------------------------------------------------------
// Kernel 2: exclusive scan of 4096 row counts -> crow[0..4096], nnz.
// Single block of 1024 threads = 32 wave32 waves. int4 vector loads.
// ---------------------------------------------------------------------------
__global__ void __launch_bounds__(1024)
scan_rows(const int* __restrict__ counts,
          int* __restrict__ crow,          // 4097 ints in d_out
          int* __restrict__ nnz_out,       // 1 int in d_out
          int* __restrict__ nnz_ws) {      // 1 int in workspace
    const int tid  = threadIdx.x;
    const int lane = tid & 31;
    const int wv   = tid >> 5;

    int4 c4 = ((const int4*)counts)[tid];            // counts[4t .. 4t+3]
    const int s0   = c4.x;
    const int s01  = s0 + c4.y;
    const int s012 = s01 + c4.z;
    const int tsum = s012 + c4.w;

    // wave32 inclusive scan of per-thread sums
    int inc = tsum;
#pragma unroll
    for (int off = 1; off < 32; off <<= 1) {
        int t = __shfl_up(inc, off, 32);
        if (lane >= off) inc += t;
    }

    __shared__ int waveTot[32];
    if (lane == 31) waveTot[wv] = inc;
    __syncthreads();
    if (wv == 0) {                                   // scan the 32 wave totals
        int v = waveTot[lane];
#pragma unroll
        for (int off = 1; off < 32; off <<= 1) {
            int t = __shfl_up(v, off, 32);
            if (lane >= off) v += t;
        }
        waveTot[lane] = v;                           // inclusive
    }
    __syncthreads();

    const int waveBase = (wv > 0) ? waveTot[wv - 1] : 0;
    const int exc = waveBase + inc - tsum;           // exclusive over threads

    crow[tid * 4 + 0] = exc;
    crow[tid * 4 + 1] = exc + s0;
    crow[tid * 4 + 2] = exc + s01;
    crow[tid * 4 + 3] = exc + s012;
    if (tid == 1023) {
        const int total = exc + tsum;                // == nnz
        crow[4096] = total;
        nnz_out[0] = total;
        nnz_ws[0]  = total;
    }
}

// ---------------------------------------------------------------------------
// Kernel 3: stable stream compaction, one block per row.
// Phase 1: compact the row into LDS (u16 cols + f32 vals, 48 KB).
// Phase 2: flush the contiguous segment [crow[row], crow[row+1]) with dense
//          full-wave stores; value bulk via async LDS->global (ASYNCcnt).
// ---------------------------------------------------------------------------
__global__ void __launch_bounds__(256)
compact_rows(const float* __restrict__ x,
             const int* __restrict__ crow,
             int* __restrict__ colI,
             float* __restrict__ vals) {
    const int row  = blockIdx.x;
    const int tid  = threadIdx.x;
    const int lane = tid & 31;
    const int wv   = tid >> 5;

    const float4* __restrict__ rowp =
        (const float4*)(x + (size_t)row * COLS);

    __shared__ unsigned short lcol[COLS];   // 16 KB (col ids fit in u16)
    __shared__ float          lval[COLS];   // 32 KB
    __shared__ int waveOff[8];
    __shared__ int iterTotal;

    int localBase = 0;                      // running compacted length

#pragma unroll 1
    for (int it = 0; it < 8; ++it) {
        const int f4idx = it * 256 + tid;            // ascending columns
        float4 v = rowp[f4idx];
        if (it < 7) {                                // prefetch next tile
            __builtin_prefetch((const void*)(rowp + f4idx + 256), 0, 3);
        }
        const int col0 = f4idx * 4;

        int m = (v.x != 0.0f) | ((v.y != 0.0f) << 1) |
                ((v.z != 0.0f) << 2) | ((v.w != 0.0f) << 3);
        const int c = __popc(m);

        // wave32 inclusive scan of per-thread counts
        int inc = c;
#pragma unroll
        for (int off = 1; off < 32; off <<= 1) {
            int t = __shfl_up(inc, off, 32);
            if (lane >= off) inc += t;
        }
        if (lane == 31) waveOff[wv] = inc;
        __syncthreads();
        if (tid == 0) {                              // exclusive scan, 8 waves
            int s = 0;
#pragma unroll
            for (int i = 0; i < 8; ++i) { int t = waveOff[i]; waveOff[i] = s; s += t; }
            iterTotal = s;
        }
        __syncthreads();

        int dest = localBase + waveOff[wv] + (inc - c);  // stable local slot
        if (m & 1) { lcol[dest] = (unsigned short)(col0);     lval[dest] = v.x; ++dest; }
        if (m & 2) { lcol[dest] = (unsigned short)(col0 + 1); lval[dest] = v.y; ++dest; }
        if (m & 4) { lcol[dest] = (unsigned short)(col0 + 2); lval[dest] = v.z; ++dest; }
        if (m & 8) { lcol[dest] = (unsigned short)(col0 + 3); lval[dest] = v.w; ++dest; }

        localBase += iterTotal;
        __syncthreads();                             // waveOff reused next iter
    }

    // ---- flush contiguous segment ----
    const int rowLen = localBase;                    // uniform across block
    const int gbase  = crow[row];                    // uniform -> s_load
    const int nFull  = rowLen & ~255;                // full-EXEC portion

#if HAVE_ASYNC_LDS_STORE
    // values: async LDS -> global (ASYNCcnt), full waves only
    for (int i = tid; i < nFull; i += 256) {
        __builtin_amdgcn_global_store_async_from_lds_b32(
            (gas_int*)(vals + gbase + i), (las_int*)&lval[i], 0, 0);
    }
#else
    for (int i = tid; i < nFull; i += 256) {
        __builtin_nontemporal_store(lval[i], &vals[gbase + i]);
    }
#endif
    // value tail (ragged EXEC) via regular NT stores
    for (int i = nFull + tid; i < rowLen; i += 256) {
        __builtin_nontemporal_store(lval[i], &vals[gbase + i]);
    }
    // cols: widen u16 -> i32, dense NT stores
    for (int i = tid; i < rowLen; i += 256) {
        __builtin_nontemporal_store((int)lcol[i], &colI[gbase + i]);
    }
#if HAVE_ASYNC_LDS_STORE
    __builtin_amdgcn_s_wait_asynccnt(0);             // LDS freed at block end
#endif
}

// ---------------------------------------------------------------------------
// Kernel 4: pad [nnz, numel) with -1 / 0.0f. Disjoint from compaction's
// [0, nnz) so the two kernels can never race. NT stores, lane-contiguous.
// ---------------------------------------------------------------------------
__global__ void __launch_bounds__(256)
fill_pad(int* __restrict__ colI, float* __restrict__ vals,
         const int* __restrict__ nnz_ws) {
    const size_t nnz  = (size_t)nnz_ws[0];
    const size_t base = (size_t)blockIdx.x * 1024u;
#pragma unroll
    for (int k = 0; k < 4; ++k) {
        const size_t i = base + (size_t)k * 256u + threadIdx.x;
        if (i >= nnz) {                              // i < NUMEL by grid size
            __builtin_nontemporal_store(-1,   &colI[i]);
            __builtin_nontemporal_store(0.0f, &vals[i]);
        }
    }
}

// ---------------------------------------------------------------------------
extern "C" void kernel_launch(void* const* d_in, const int* in_sizes, int n_in,
                              void* d_out, int out_size, void* d_ws, size_t ws_size,
                              hipStream_t stream) {
    (void)in_sizes; (void)n_in; (void)out_size; (void)ws_size;

    const float* x = (const float*)d_in[0];

    // d_out layout: crow[4097] | col_indices[NUMEL] (int bits) |
    //               values[NUMEL] (f32) | nnz[1] (int bits)
    int*   crow    = (int*)d_out;
    int*   colI    = crow + (ROWS + 1);
    float* vals    = (float*)d_out + (ROWS + 1) + NUMEL;
    int*   nnz_out = (int*)d_out + (ROWS + 1) + 2 * NUMEL;

    // workspace: per-row counts + nnz scalar
    int* counts = (int*)d_ws;
    int* nnz_ws = counts + ROWS;

    // count -> scan -> compact (x re-read hits 192MB L2) -> fill padding
    count_rows<<<ROWS, 256, 0, stream>>>(x, counts);
    scan_rows<<<1, 1024, 0, stream>>>(counts, crow, nnz_out, nnz_ws);
    compact_rows<<<ROWS, 256, 0, stream>>>(x, crow, colI, vals);
    fill_pad<<<(unsigned)(NUMEL / 1024), 256, 0, stream>>>(colI, vals, nnz_ws);
}